// Def_Block_SR_12807592477451
// MI455X (gfx1250) — compile-verified
//
#include <hip/hip_runtime.h>
#include <math.h>

#define HW      128
#define CIN     64
#define KTOT    576          // 64 channels * 9 taps
#define NSTEPS  144          // KTOT / 4 (K=4 per f32 WMMA)
#define NPX     32           // pixels per block tile (2 x 16-wide N tiles)
#define NS      17           // LDS panel column stride (pad -> conflict-free)
#define PANEL   (KTOT * NS)  // 9792 floats per 16-pixel panel
#define PS      (PANEL + 16) // panel-to-panel stride; +16 dwords keeps banks disjoint
#define OFFST   20           // offset buffer channel stride (18 used)

typedef float v2f __attribute__((ext_vector_type(2)));
typedef float v4f __attribute__((ext_vector_type(4)));
typedef float v8f __attribute__((ext_vector_type(8)));

// panel element (kk, pixel-col c in 0..31) -> LDS float index
#define BIDX(kk, c) ((((c) >> 4) * PS) + (kk) * NS + ((c) & 15))

// ---------------------------------------------------------------------------
// WMMA GEMM over two 16-col LDS panels with A-fragment reuse.
// A fragment (pre-swizzled in global):
//   lanes 0-15 : M = lane,    K = {4s+0, 4s+1}
//   lanes 16-31: M = lane-16, K = {4s+2, 4s+3}
// B fragment from panel q: lanes 0-15: N = lane, K pair {4s+0,4s+1};
//                          lanes 16-31: N = lane-16, K pair {4s+2,4s+3}
// ---------------------------------------------------------------------------
__device__ __forceinline__ void gemm_panel2(const float* __restrict__ pw,
                                            const float* __restrict__ Bp,
                                            int lane, v8f& acc0, v8f& acc1) {
  const int n    = lane & 15;
  const int kofs = (lane >> 4) << 1;
#pragma unroll 4
  for (int s = 0; s < NSTEPS; ++s) {
    v2f a = *(const v2f*)(pw + (((s << 5) + lane) << 1));
    const int kk = (s << 2) + kofs;
    v2f b0, b1;
    b0.x = Bp[kk * NS + n];
    b0.y = Bp[(kk + 1) * NS + n];
    b1.x = Bp[PS + kk * NS + n];
    b1.y = Bp[PS + (kk + 1) * NS + n];
    acc0 = __builtin_amdgcn_wmma_f32_16x16x4_f32(false, a, false, b0,
                                                 (short)0, acc0, false, false);
    acc1 = __builtin_amdgcn_wmma_f32_16x16x4_f32(false, a, false, b1,
                                                 (short)0, acc1, false, false);
  }
}

// ---------------------------------------------------------------------------
// Prep: fold conv bias + BN into per-channel affine (scale, shift).
// ---------------------------------------------------------------------------
__global__ void prep_params_kernel(const float* __restrict__ c1b,
                                   const float* __restrict__ g1, const float* __restrict__ b1,
                                   const float* __restrict__ m1, const float* __restrict__ v1,
                                   const float* __restrict__ dcb,
                                   const float* __restrict__ g2, const float* __restrict__ b2,
                                   const float* __restrict__ m2, const float* __restrict__ v2,
                                   float* __restrict__ scale1, float* __restrict__ shift1,
                                   float* __restrict__ scale2, float* __restrict__ shift2) {
  int c = threadIdx.x;
  if (c < 64) {
    float s = g1[c] * rsqrtf(v1[c] + 1e-5f);
    scale1[c] = s;
    shift1[c] = (c1b[c] - m1[c]) * s + b1[c];
    float t = g2[c] * rsqrtf(v2[c] + 1e-5f);
    scale2[c] = t;
    shift2[c] = (dcb[c] - m2[c]) * t + b2[c];
  }
}

// ---------------------------------------------------------------------------
// Prep: swizzle OIHW weights into per-lane WMMA A-fragment order so the GEMM
// loop does one coalesced b64 load per step.
// ---------------------------------------------------------------------------
__global__ void prep_weights_kernel(const float* __restrict__ w,
                                    float* __restrict__ dst,
                                    int cout, int total) {
  int idx = blockIdx.x * blockDim.x + threadIdx.x;
  if (idx >= total) return;
  int lane = idx & 31;
  int s    = (idx >> 5) % NSTEPS;
  int t    = (idx >> 5) / NSTEPS;
  int o    = (t << 4) + (lane & 15);
  int kk   = (s << 2) + ((lane >> 4) << 1);
  float a0 = 0.f, a1 = 0.f;
  if (o < cout) {
    a0 = w[o * KTOT + kk];
    a1 = w[o * KTOT + kk + 1];
  }
  dst[(idx << 1) + 0] = a0;
  dst[(idx << 1) + 1] = a1;
}

// ---------------------------------------------------------------------------
// conv1 + BN1 + ReLU.  Input x NCHW, output out1 NHWC (for fast sampling).
// Block = 32 pixels in one row, 128 threads = 4 waves x (16 ch x 32 px).
// ---------------------------------------------------------------------------
__global__ __launch_bounds__(128) void conv1_kernel(
    const float* __restrict__ x, const float* __restrict__ pw,
    const float* __restrict__ scale, const float* __restrict__ shift,
    float* __restrict__ out1) {
  __shared__ float Bp[2 * PS];
  const int bid = blockIdx.x;
  const int img = bid >> 9;                 // 128 rows * 4 tiles = 512 tiles/img
  const int rem = bid & 511;
  const int py  = rem >> 2;
  const int px0 = (rem & 3) << 5;
  const int tid = threadIdx.x;

  {  // im2col fill: Bp(kk = c*9 + ky*3 + kx, n) = x[c][py+ky-1][px0+n+kx-1]
    const int n = tid & 31;
    const float* xi = x + (size_t)img * CIN * HW * HW;
    for (int kk = tid >> 5; kk < KTOT; kk += 4) {
      int c  = kk / 9;
      int r  = kk - c * 9;
      int ky = r / 3;
      int kx = r - ky * 3;
      int yy = py + ky - 1;
      int xx = px0 + n + kx - 1;
      float v = 0.f;
      if (yy >= 0 && yy < HW && xx >= 0 && xx < HW)
        v = xi[(c * HW + yy) * HW + xx];
      Bp[BIDX(kk, n)] = v;
    }
  }
  __syncthreads();

  const int lane = tid & 31;
  const int mt   = tid >> 5;  // output-channel tile 0..3
  v8f acc0 = {0.f, 0.f, 0.f, 0.f, 0.f, 0.f, 0.f, 0.f};
  v8f acc1 = acc0;
  gemm_panel2(pw + mt * (NSTEPS * 64), Bp, lane, acc0, acc1);

  const int ch0 = (mt << 4) + ((lane >> 4) << 3);
  const int n   = lane & 15;
  const size_t base = (size_t)img * (HW * HW) + py * HW + px0 + n;
  float* o0 = out1 + base * 64 + ch0;
  float* o1 = out1 + (base + 16) * 64 + ch0;
#pragma unroll
  for (int r = 0; r < 8; ++r) {
    float s = scale[ch0 + r], sh = shift[ch0 + r];
    float v0 = acc0[r] * s + sh;
    float v1 = acc1[r] * s + sh;
    o0[r] = v0 > 0.f ? v0 : 0.f;
    o1[r] = v1 > 0.f ? v1 : 0.f;
  }
}

// ---------------------------------------------------------------------------
// Offset conv: 64->18 channels over out1 (NHWC).  64 threads = 2 waves.
// Output offp[pixel*20 + c], c<18, channel c=2k dy, c=2k+1 dx.
// ---------------------------------------------------------------------------
__global__ __launch_bounds__(64) void offconv_kernel(
    const float* __restrict__ out1, const float* __restrict__ pw,
    const float* __restrict__ offb, float* __restrict__ offp) {
  __shared__ float Bp[2 * PS];
  const int bid = blockIdx.x;
  const int img = bid >> 9;
  const int rem = bid & 511;
  const int py  = rem >> 2;
  const int px0 = (rem & 3) << 5;
  const int tid = threadIdx.x;

  {  // fill from NHWC: each 16-lane group handles sites (k, p), 4 ch / lane
    const int g = tid >> 4;    // 0..3
    const int n = tid & 15;
    const float* o1 = out1 + (size_t)img * HW * HW * 64;
    for (int site = g; site < 288; site += 4) {
      int k  = site >> 5;      // tap 0..8
      int p  = site & 31;      // pixel 0..31
      int yy = py + k / 3 - 1;
      int xx = px0 + p + (k % 3) - 1;
      v4f v = {0.f, 0.f, 0.f, 0.f};
      if (yy >= 0 && yy < HW && xx >= 0 && xx < HW)
        v = *(const v4f*)(o1 + ((size_t)yy * HW + xx) * 64 + (n << 2));
#pragma unroll
      for (int j = 0; j < 4; ++j)
        Bp[BIDX(((n << 2) + j) * 9 + k, p)] = v[j];
    }
  }
  __syncthreads();

  const int lane = tid & 31;
  const int mt   = tid >> 5;  // 0..1
  v8f acc0 = {0.f, 0.f, 0.f, 0.f, 0.f, 0.f, 0.f, 0.f};
  v8f acc1 = acc0;
  gemm_panel2(pw + mt * (NSTEPS * 64), Bp, lane, acc0, acc1);

  const int ch0 = (mt << 4) + ((lane >> 4) << 3);
  const int n   = lane & 15;
  const size_t pix = (size_t)img * (HW * HW) + py * HW + px0 + n;
#pragma unroll
  for (int r = 0; r < 8; ++r) {
    int ch = ch0 + r;
    if (ch < 18) {
      float b = offb[ch];
      offp[pix * OFFST + ch]        = acc0[r] + b;
      offp[(pix + 16) * OFFST + ch] = acc1[r] + b;
    }
  }
}

// ---------------------------------------------------------------------------
// Deformable conv + BN2 + identity add + ReLU.  Bilinear sampler fills the
// LDS panels from NHWC out1; identity x and output are NCHW.
// ---------------------------------------------------------------------------
__global__ __launch_bounds__(128) void deform_kernel(
    const float* __restrict__ out1, const float* __restrict__ offp,
    const float* __restrict__ pw, const float* __restrict__ scale,
    const float* __restrict__ shift, const float* __restrict__ x,
    float* __restrict__ out) {
  __shared__ float Bp[2 * PS];
  const int bid = blockIdx.x;
  const int img = bid >> 9;
  const int rem = bid & 511;
  const int py  = rem >> 2;
  const int px0 = (rem & 3) << 5;
  const int tid = threadIdx.x;

  {  // bilinear sample fill: each 16-lane group = one (tap k, pixel p) site
    const int g = tid >> 4;    // 0..7
    const int n = tid & 15;
    const float* o1 = out1 + (size_t)img * HW * HW * 64;
    for (int site = g; site < 288; site += 8) {
      int k = site >> 5;       // tap 0..8
      int p = site & 31;       // pixel 0..31
      const size_t pix = (size_t)img * (HW * HW) + py * HW + px0 + p;
      float dy = offp[pix * OFFST + 2 * k];
      float dx = offp[pix * OFFST + 2 * k + 1];
      float ys = (float)(py + k / 3 - 1) + dy;
      float xs = (float)(px0 + p + (k % 3) - 1) + dx;
      float y0f = floorf(ys), x0f = floorf(xs);
      float wy = ys - y0f, wx = xs - x0f;
      int y0 = (int)y0f, x0 = (int)x0f;
      v4f acc4 = {0.f, 0.f, 0.f, 0.f};
      const float w00 = (1.f - wy) * (1.f - wx);
      const float w01 = (1.f - wy) * wx;
      const float w10 = wy * (1.f - wx);
      const float w11 = wy * wx;
#pragma unroll
      for (int c4 = 0; c4 < 4; ++c4) {
        int yy = y0 + (c4 >> 1);
        int xx = x0 + (c4 & 1);
        float wgt = (c4 == 0) ? w00 : (c4 == 1) ? w01 : (c4 == 2) ? w10 : w11;
        if (yy >= 0 && yy < HW && xx >= 0 && xx < HW) {
          v4f v = *(const v4f*)(o1 + ((size_t)yy * HW + xx) * 64 + (n << 2));
          acc4 += v * wgt;
        }
      }
#pragma unroll
      for (int j = 0; j < 4; ++j)
        Bp[BIDX(((n << 2) + j) * 9 + k, p)] = acc4[j];
    }
  }
  __syncthreads();

  const int lane = tid & 31;
  const int mt   = tid >> 5;  // 0..3
  v8f acc0 = {0.f, 0.f, 0.f, 0.f, 0.f, 0.f, 0.f, 0.f};
  v8f acc1 = acc0;
  gemm_panel2(pw + mt * (NSTEPS * 64), Bp, lane, acc0, acc1);

  const int ch0 = (mt << 4) + ((lane >> 4) << 3);
  const int n   = lane & 15;
  const int yx  = py * HW + px0 + n;
#pragma unroll
  for (int r = 0; r < 8; ++r) {
    int ch = ch0 + r;
    size_t idx = ((size_t)(img * 64 + ch)) * (HW * HW) + yx;
    float s = scale[ch], sh = shift[ch];
    float v0 = acc0[r] * s + sh + x[idx];
    float v1 = acc1[r] * s + sh + x[idx + 16];
    out[idx]      = v0 > 0.f ? v0 : 0.f;
    out[idx + 16] = v1 > 0.f ? v1 : 0.f;
  }
}

// ---------------------------------------------------------------------------
extern "C" void kernel_launch(void* const* d_in, const int* in_sizes, int n_in,
                              void* d_out, int out_size, void* d_ws, size_t ws_size,
                              hipStream_t stream) {
  const float* x    = (const float*)d_in[0];
  const float* c1w  = (const float*)d_in[1];
  const float* c1b  = (const float*)d_in[2];
  const float* g1   = (const float*)d_in[3];
  const float* b1   = (const float*)d_in[4];
  const float* m1   = (const float*)d_in[5];
  const float* v1   = (const float*)d_in[6];
  const float* offw = (const float*)d_in[7];
  const float* offb = (const float*)d_in[8];
  const float* dcw  = (const float*)d_in[9];
  const float* dcb  = (const float*)d_in[10];
  const float* g2   = (const float*)d_in[11];
  const float* b2   = (const float*)d_in[12];
  const float* m2   = (const float*)d_in[13];
  const float* v2   = (const float*)d_in[14];
  float* out = (float*)d_out;

  float* ws = (float*)d_ws;
  size_t o = 0;
  float* out1 = ws + o;  o += (size_t)8 * HW * HW * 64;     // NHWC activations
  float* offp = ws + o;  o += (size_t)8 * HW * HW * OFFST;  // offsets, stride 20
  float* pw1  = ws + o;  o += (size_t)4 * NSTEPS * 64;      // swizzled weights
  float* pwo  = ws + o;  o += (size_t)2 * NSTEPS * 64;
  float* pw2  = ws + o;  o += (size_t)4 * NSTEPS * 64;
  float* scale1 = ws + o; o += 64;
  float* shift1 = ws + o; o += 64;
  float* scale2 = ws + o; o += 64;
  float* shift2 = ws + o; o += 64;

  prep_params_kernel<<<1, 64, 0, stream>>>(c1b, g1, b1, m1, v1, dcb, g2, b2, m2, v2,
                                           scale1, shift1, scale2, shift2);
  {
    int t1 = 4 * NSTEPS * 32, to = 2 * NSTEPS * 32;
    prep_weights_kernel<<<(t1 + 255) / 256, 256, 0, stream>>>(c1w, pw1, 64, t1);
    prep_weights_kernel<<<(to + 255) / 256, 256, 0, stream>>>(offw, pwo, 18, to);
    prep_weights_kernel<<<(t1 + 255) / 256, 256, 0, stream>>>(dcw, pw2, 64, t1);
  }

  const int nblocks = 8 * HW * (HW / NPX);  // 4096
  conv1_kernel<<<nblocks, 128, 0, stream>>>(x, pw1, scale1, shift1, out1);
  offconv_kernel<<<nblocks, 64, 0, stream>>>(out1, pwo, offb, offp);
  deform_kernel<<<nblocks, 128, 0, stream>>>(out1, offp, pw2, scale2, shift2, x, out);
}